// GraphAttentionLayer_18433999635189
// MI455X (gfx1250) — compile-verified
//
#include <hip/hip_runtime.h>

typedef __attribute__((ext_vector_type(2))) float v2f;
typedef __attribute__((ext_vector_type(8))) float v8f;

#define IN_CH 128
#define LDW   130   // LDS row pitch (floats): even (8B-aligned float2) + conflict-free

// ---------------------------------------------------------------------------
// Kernel 1: h = x @ W^T via V_WMMA_F32_16X16X4_F32.
// Block = 256 threads (8 waves). Each wave owns 16 node rows and the full 128
// output channels (8 accumulator tiles of 16x16). W (128x128) staged in LDS.
// ---------------------------------------------------------------------------
__global__ __launch_bounds__(256) void gat_gemm_wmma(const float* __restrict__ x,
                                                     const float* __restrict__ W,
                                                     float* __restrict__ h, int N)
{
    __shared__ float Wlds[128 * LDW];
    for (int i = threadIdx.x; i < 128 * 128; i += 256) {
        int o = i >> 7, k = i & 127;
        Wlds[o * LDW + k] = W[i];
    }
    __syncthreads();

    const int lane = threadIdx.x & 31;
    const int wave = threadIdx.x >> 5;
    const int nodeBase = blockIdx.x * 128 + wave * 16;
    const int hi  = lane >> 4;   // 0: lanes 0-15 (K=k0,k0+1), 1: lanes 16-31 (K=k0+2,k0+3)
    const int l16 = lane & 15;

    int arow = nodeBase + l16;
    if (arow >= N) arow = N - 1;            // clamp loads; stores are guarded

    v8f acc[8] = {};
    const float* xrow = x + (size_t)arow * IN_CH + hi * 2;

    for (int k0 = 0; k0 < IN_CH; k0 += 4) {
        const float2 a2 = *(const float2*)(xrow + k0);
        v2f a; a.x = a2.x; a.y = a2.y;
        #pragma unroll
        for (int t = 0; t < 8; ++t) {
            const int och = t * 16 + l16;   // B[k][och] = W[och][k]
            const float2 b2 = *(const float2*)(&Wlds[och * LDW + k0 + hi * 2]);
            v2f b; b.x = b2.x; b.y = b2.y;
            acc[t] = __builtin_amdgcn_wmma_f32_16x16x4_f32(
                false, a, false, b, (short)0, acc[t], false, false);
        }
    }

    // C layout: vgpr r -> row M = r + hi*8, col N = l16
    #pragma unroll
    for (int t = 0; t < 8; ++t) {
        const int col = t * 16 + l16;
        #pragma unroll
        for (int r = 0; r < 8; ++r) {
            const int row = nodeBase + hi * 8 + r;
            if (row < N) h[(size_t)row * IN_CH + col] = acc[t][r];
        }
    }
}

// ---------------------------------------------------------------------------
// Kernel 2: per-node logits  s_src[n,h] = sum_c h[n,h,c]*a_src[h,c]  (same for dst)
// Block = 128 threads = 4 waves; wave w handles head w; wave-level shuffle reduce.
// ---------------------------------------------------------------------------
__global__ __launch_bounds__(128) void gat_scores(const float* __restrict__ h,
                                                  const float* __restrict__ a_src,
                                                  const float* __restrict__ a_dst,
                                                  float* __restrict__ s_src,
                                                  float* __restrict__ s_dst, int N)
{
    const int n = blockIdx.x;
    const int t = threadIdx.x;             // 0..127 : head = t>>5, c = t&31
    const float hv = h[(size_t)n * 128 + t];
    float ss = hv * a_src[t];
    float sd = hv * a_dst[t];
    #pragma unroll
    for (int off = 16; off > 0; off >>= 1) {
        ss += __shfl_down(ss, off, 32);
        sd += __shfl_down(sd, off, 32);
    }
    if ((t & 31) == 0) {
        s_src[n * 4 + (t >> 5)] = ss;
        s_dst[n * 4 + (t >> 5)] = sd;
    }
}

// ---------------------------------------------------------------------------
// Kernel 3: segment max of alpha over dst. Init is 0 (reference includes the
// zero init in its max), so only alpha>0 matters -> uint-bit atomicMax is exact.
// ---------------------------------------------------------------------------
__global__ void gat_edge_max(const long long* __restrict__ ei,
                             const float* __restrict__ ew,
                             const float* __restrict__ s_src,
                             const float* __restrict__ s_dst,
                             unsigned int* __restrict__ amax_u, int E)
{
    const int e = blockIdx.x * blockDim.x + threadIdx.x;
    if (e >= E) return;
    const int src = (int)ei[e];
    const int dst = (int)ei[(size_t)E + e];
    const float w = ew[e];
    #pragma unroll
    for (int hd = 0; hd < 4; ++hd) {
        const float z = s_src[src * 4 + hd] + s_dst[dst * 4 + hd];
        const float a = (z > 0.0f ? z : 0.2f * z) * w;
        if (a > 0.0f) atomicMax(&amax_u[dst * 4 + hd], __float_as_uint(a));
    }
}

// ---------------------------------------------------------------------------
// Kernel 4: per-edge p = exp(alpha - amax[dst]); accumulate unnormalized
// numerator out[dst] += p*h[src] and denominator asum[dst] += p.
// 128 threads per edge -> coalesced 512B gather/atomic rows; L2-resident.
// ---------------------------------------------------------------------------
__global__ __launch_bounds__(256) void gat_edge_agg(const long long* __restrict__ ei,
                                                    const float* __restrict__ ew,
                                                    const float* __restrict__ s_src,
                                                    const float* __restrict__ s_dst,
                                                    const float* __restrict__ amax,
                                                    const float* __restrict__ h,
                                                    float* __restrict__ asum,
                                                    float* __restrict__ out, int E)
{
    const int eloc = threadIdx.x >> 7;          // 2 edges per 256-thread block
    const int li   = threadIdx.x & 127;         // channel within node row
    const int e    = blockIdx.x * 2 + eloc;
    if (e >= E) return;
    const int hd  = li >> 5;
    const int src = (int)ei[e];
    const int dst = (int)ei[(size_t)E + e];
    const float w = ew[e];
    const float z = s_src[src * 4 + hd] + s_dst[dst * 4 + hd];
    const float a = (z > 0.0f ? z : 0.2f * z) * w;
    const float p = __expf(a - amax[dst * 4 + hd]);
    if ((li & 31) == 0) atomicAdd(&asum[dst * 4 + hd], p);
    const float hv = h[(size_t)src * 128 + li];
    atomicAdd(&out[(size_t)dst * 128 + li], p * hv);
}

// ---------------------------------------------------------------------------
// Kernel 5: normalize out by softmax denominator.
// ---------------------------------------------------------------------------
__global__ void gat_norm(float* __restrict__ out, const float* __restrict__ asum, int N)
{
    const int i = blockIdx.x * blockDim.x + threadIdx.x;
    if (i >= N * 128) return;
    const int n  = i >> 7;
    const int hd = (i >> 5) & 3;
    out[i] = out[i] / (asum[n * 4 + hd] + 1e-8f);
}

extern "C" void kernel_launch(void* const* d_in, const int* in_sizes, int n_in,
                              void* d_out, int out_size, void* d_ws, size_t ws_size,
                              hipStream_t stream)
{
    const float*     x     = (const float*)d_in[0];
    const long long* ei    = (const long long*)d_in[1];   // int64 (2,E)
    const float*     ew    = (const float*)d_in[2];
    const float*     W     = (const float*)d_in[3];
    const float*     a_src = (const float*)d_in[4];
    const float*     a_dst = (const float*)d_in[5];
    const int N = in_sizes[0] / 128;
    const int E = in_sizes[1] / 2;
    float* out = (float*)d_out;

    // workspace layout (floats): h[N*128] | s_src[N*4] | s_dst[N*4] | amax[N*4] | asum[N*4]
    float* h    = (float*)d_ws;
    float* ssrc = h    + (size_t)N * 128;
    float* sdst = ssrc + (size_t)N * 4;
    float* amax = sdst + (size_t)N * 4;
    float* asum = amax + (size_t)N * 4;

    // amax & asum are contiguous -> one async memset; zero the output accumulator
    hipMemsetAsync(amax, 0, (size_t)N * 8 * sizeof(float), stream);
    hipMemsetAsync(out, 0, (size_t)out_size * sizeof(float), stream);

    gat_gemm_wmma<<<(N + 127) / 128, 256, 0, stream>>>(x, W, h, N);
    gat_scores  <<<N, 128, 0, stream>>>(h, a_src, a_dst, ssrc, sdst, N);
    gat_edge_max<<<(E + 255) / 256, 256, 0, stream>>>(ei, ew, ssrc, sdst,
                                                      (unsigned int*)amax, E);
    gat_edge_agg<<<(E + 1) / 2, 256, 0, stream>>>(ei, ew, ssrc, sdst, amax, h, asum, out, E);
    gat_norm    <<<((N * 128) + 255) / 256, 256, 0, stream>>>(out, asum, N);
}